// Level1Model_64269890618093
// MI455X (gfx1250) — compile-verified
//
#include <hip/hip_runtime.h>
#include <hip/hip_bf16.h>
#include <math.h>

typedef __attribute__((ext_vector_type(2))) float v2f;
typedef __attribute__((ext_vector_type(8))) float v8f;

#define N_NODES    50000
#define N_EDGES    800000
#define IN_DIM     64
#define HID        128
#define NUM_GRAPHS 256

// ---------------------------------------------------------------------------
// Weight pre-swizzle: W[K x 128] row-major  ->  WMMA B-fragment order.
// For k-step ks = k>>2, kh = (k>>1)&1: the pair (W[4ks+2kh][col], W[4ks+2kh+1][col])
// lives contiguously at float index ((ks*128 + col)*2 + kh)*2.  A lane then
// fetches its whole B fragment as ONE global_load_b64 whose per-iteration
// offset is the compile-time constant ks*2048 bytes (24-bit IOFFSET).
// ---------------------------------------------------------------------------
__global__ void swizzle_w_kernel(const float* __restrict__ W,
                                 float* __restrict__ Wsw, int KC) {
  const int idx = blockIdx.x * blockDim.x + threadIdx.x;
  if (idx >= KC) return;
  const int k   = idx >> 7;    // row (C == 128)
  const int col = idx & 127;
  const int fidx = (((k >> 2) * 128 + col) << 2) + (((k >> 1) & 1) << 1) + (k & 1);
  Wsw[fidx] = W[idx];
}

// ---------------------------------------------------------------------------
// WMMA f32 GEMM:  out[N x 128] = act( A[N x K] (+ Aadd) @ W[K x 128] + bias )
//
// blockDim = 256 (8 waves); block computes 16 rows x 128 cols, wave = one
// 16x16 tile via chained V_WMMA_F32_16X16X4_F32 (K/4 matrix ops).
//
// A tile (16 x K, with fused h+agg add) is staged ONCE per block into LDS in
// A-fragment order and shared by all 8 waves.  Per-ks block is padded to 68
// floats so fragment ds_load_b64s are bank-conflict-free (lanes 0-15 -> banks
// 0,4,..,60; lanes 16-31 -> banks 2,6,..,62) and so are the staging stores.
//
// Inner loop per ks: 1x ds_load_b64 (A frag) + 1x global_load_b64 (B frag,
// immediate offset) + 1x v_wmma_f32_16x16x4_f32.  All dims are exact tile
// multiples so EXEC is all-1s (WMMA requirement).
// ---------------------------------------------------------------------------
template<int K, bool FUSE_ADD, bool RELU>
__global__ void __launch_bounds__(256)
gemm_wmma_f32(const float* __restrict__ A,
              const float* __restrict__ Aadd,
              const float* __restrict__ Wsw,   // pre-swizzled weights
              const float* __restrict__ bias,
              float* __restrict__ out) {
  constexpr int C   = HID;      // 128 for every GEMM in this model
  constexpr int NKS = K / 4;    // number of 16x16x4 steps
  constexpr int AS  = 68;       // padded floats per ks-block in LDS (64 + 4)
  constexpr int KB  = (K == 64) ? 6 : (K == 128) ? 7 : 8;
  __shared__ float sAf[NKS * AS];

  const int tid     = threadIdx.x;
  const int rowBase = blockIdx.x * 16;

  // ---- stage 16 x K A-tile (fused add) into LDS in fragment order ----
  for (int idx = tid; idx < 16 * K; idx += 256) {
    const int r = idx >> KB;        // row within tile
    const int k = idx & (K - 1);    // coalesced along k
    float v = A[(size_t)(rowBase + r) * K + k];
    if (FUSE_ADD) v += Aadd[(size_t)(rowBase + r) * K + k];
    const int fidx = (k >> 2) * AS + (r << 2) + (((k >> 1) & 1) << 1) + (k & 1);
    sAf[fidx] = v;
  }
  __syncthreads();

  const int lane = tid & 31;
  const int wave = tid >> 5;
  const int mrow = lane & 15;     // M row (A) / N col (B,C,D)
  const int kh   = lane >> 4;     // which K-pair of the 4-wide step
  const int col  = wave * 16 + mrow;

  const float* __restrict__ aPtr = sAf + (mrow << 2) + (kh << 1);
  const float* __restrict__ bPtr = Wsw + ((size_t)col << 2) + (kh << 1);

  v8f acc = {};
#pragma unroll 8
  for (int ks = 0; ks < NKS; ++ks) {
    const v2f a = *(const v2f*)(aPtr + ks * AS);        // ds_load_b64, imm offs
    const v2f b = *(const v2f*)(bPtr + ks * (C * 4));   // global_load_b64, imm offs
    acc = __builtin_amdgcn_wmma_f32_16x16x4_f32(
        /*neg_a=*/false, a, /*neg_b=*/false, b,
        /*c_mod=*/(short)0, acc, /*reuse_a=*/false, /*reuse_b=*/false);
  }

  const float bs   = bias[col];
  const int   rOff = kh << 3;     // 0 or 8
#pragma unroll
  for (int i = 0; i < 8; ++i) {
    float v = acc[i] + bs;
    if (RELU) v = fmaxf(v, 0.0f);
    out[(size_t)(rowBase + rOff + i) * C + col] = v;
  }
}

// ---------------------------------------------------------------------------
// Edge scatter-add:  agg[dst[e]][:] += h[src[e]][:]  (f32 atomics; node matrix
// is 25.6 MB so the atomics resolve in the 192 MB L2, not HBM).
// ---------------------------------------------------------------------------
__global__ void scatter_add_kernel(const float* __restrict__ h,
                                   const int* __restrict__ src,
                                   const int* __restrict__ dst,
                                   float* __restrict__ agg,
                                   int nEdges, int D) {
  const int t   = blockIdx.x * blockDim.x + threadIdx.x;
  const int tpe = D >> 2;
  const int e   = t / tpe;
  if (e >= nEdges) return;
  const int c = (t - e * tpe) << 2;
  const int s = src[e];
  const int d = dst[e];
  const float4 v = *reinterpret_cast<const float4*>(h + (size_t)s * D + c);
  float* p = agg + (size_t)d * D + c;
  atomicAdd(p + 0, v.x);
  atomicAdd(p + 1, v.y);
  atomicAdd(p + 2, v.z);
  atomicAdd(p + 3, v.w);
}

// ---------------------------------------------------------------------------
// Graph readout: per-graph sum, max (bitwise int max is valid: h >= 0 after
// ReLU, and nonneg floats are order-isomorphic to their int bit patterns),
// and node count.
// ---------------------------------------------------------------------------
__global__ void readout_kernel(const float* __restrict__ h,
                               const int* __restrict__ batch,
                               float* __restrict__ ssum,
                               int* __restrict__ smax_bits,
                               float* __restrict__ cnt,
                               int N, int D) {
  const int t = blockIdx.x * blockDim.x + threadIdx.x;
  if (t >= N * D) return;
  const int n  = t / D;
  const int dd = t - n * D;
  const int g  = batch[n];
  const float v = h[t];
  atomicAdd(&ssum[(size_t)g * D + dd], v);
  atomicMax(&smax_bits[(size_t)g * D + dd], __float_as_int(v));
  if (dd == 0) atomicAdd(&cnt[g], 1.0f);
}

// graph_repr[g] = concat(mean_pool[g], max_pool[g])  -> [G x 2D]
__global__ void build_repr_kernel(const float* __restrict__ ssum,
                                  const int* __restrict__ smax_bits,
                                  const float* __restrict__ cnt,
                                  float* __restrict__ repr,
                                  int G, int D) {
  const int t = blockIdx.x * blockDim.x + threadIdx.x;
  if (t >= G * 2 * D) return;
  const int g = t / (2 * D);
  const int c = t - g * 2 * D;
  const float n = cnt[g];
  float val;
  if (c < D) {
    val = ssum[(size_t)g * D + c] / fmaxf(n, 1.0f);
  } else {
    val = (n > 0.0f) ? __int_as_float(smax_bits[(size_t)g * D + (c - D)]) : 0.0f;
  }
  repr[t] = val;
}

// logits = hh @ w2 + b2 ; score = sigmoid(logits)   (one thread per graph)
__global__ void head_out_kernel(const float* __restrict__ hh,
                                const float* __restrict__ w2,
                                const float* __restrict__ b2,
                                float* __restrict__ out,
                                int G, int D) {
  const int g = blockIdx.x * blockDim.x + threadIdx.x;
  if (g >= G) return;
  float acc = b2[0];
  for (int j = 0; j < D; ++j) acc += hh[(size_t)g * D + j] * w2[j];
  out[g] = 1.0f / (1.0f + expf(-acc));
}

// ---------------------------------------------------------------------------
extern "C" void kernel_launch(void* const* d_in, const int* in_sizes, int n_in,
                              void* d_out, int out_size, void* d_ws, size_t ws_size,
                              hipStream_t stream) {
  const float* x     = (const float*)d_in[0];
  const int*   ei    = (const int*)d_in[1];
  const int*   batch = (const int*)d_in[2];
  const float* g_w1[3] = {(const float*)d_in[3],  (const float*)d_in[7],  (const float*)d_in[11]};
  const float* g_b1[3] = {(const float*)d_in[4],  (const float*)d_in[8],  (const float*)d_in[12]};
  const float* g_w2[3] = {(const float*)d_in[5],  (const float*)d_in[9],  (const float*)d_in[13]};
  const float* g_b2[3] = {(const float*)d_in[6],  (const float*)d_in[10], (const float*)d_in[14]};
  const float* head_w1 = (const float*)d_in[15];
  const float* head_b1 = (const float*)d_in[16];
  const float* head_w2 = (const float*)d_in[17];
  const float* head_b2 = (const float*)d_in[18];
  float* out = (float*)d_out;

  const int* src = ei;            // edge_index[0]
  const int* dst = ei + N_EDGES;  // edge_index[1]

  // --- workspace layout (all fp32) ---
  char* w = (char*)d_ws;
  float* h    = (float*)w; w += (size_t)N_NODES * HID * 4;        // 25.6 MB
  float* agg  = (float*)w; w += (size_t)N_NODES * HID * 4;        // 25.6 MB
  float* tbuf = (float*)w; w += (size_t)N_NODES * HID * 4;        // 25.6 MB
  float* ssum = (float*)w; w += (size_t)NUM_GRAPHS * HID * 4;
  int*   smax = (int*)w;   w += (size_t)NUM_GRAPHS * HID * 4;
  float* cnt  = (float*)w; w += (size_t)NUM_GRAPHS * 4;
  float* repr = (float*)w; w += (size_t)NUM_GRAPHS * 2 * HID * 4;
  float* hh   = (float*)w; w += (size_t)NUM_GRAPHS * HID * 4;
  // pre-swizzled weights (B-fragment order)
  float* wsw1[3];
  float* wsw2[3];
  for (int l = 0; l < 3; ++l) {
    wsw1[l] = (float*)w; w += (size_t)((l == 0) ? IN_DIM : HID) * HID * 4;
    wsw2[l] = (float*)w; w += (size_t)HID * HID * 4;
  }
  float* wswh = (float*)w; w += (size_t)(2 * HID) * HID * 4;
  (void)ws_size; (void)in_sizes; (void)n_in; (void)out_size;

  // ---- swizzle all weight matrices into WMMA B-fragment order ----
  for (int l = 0; l < 3; ++l) {
    const int kc1 = ((l == 0) ? IN_DIM : HID) * HID;
    swizzle_w_kernel<<<(kc1 + 255) / 256, 256, 0, stream>>>(g_w1[l], wsw1[l], kc1);
    swizzle_w_kernel<<<(HID * HID + 255) / 256, 256, 0, stream>>>(g_w2[l], wsw2[l], HID * HID);
  }
  swizzle_w_kernel<<<(2 * HID * HID + 255) / 256, 256, 0, stream>>>(head_w1, wswh, 2 * HID * HID);

  const int rowBlocks = N_NODES / 16;  // 3125 exactly

  // ---- GIN layer 0 (din = 64) ----
  hipMemsetAsync(agg, 0, (size_t)N_NODES * IN_DIM * 4, stream);
  {
    const int threads = N_EDGES * (IN_DIM / 4);
    scatter_add_kernel<<<(threads + 255) / 256, 256, 0, stream>>>(
        x, src, dst, agg, N_EDGES, IN_DIM);
  }
  gemm_wmma_f32<IN_DIM, true, true><<<rowBlocks, 256, 0, stream>>>(
      x, agg, wsw1[0], g_b1[0], tbuf);
  gemm_wmma_f32<HID, false, true><<<rowBlocks, 256, 0, stream>>>(
      tbuf, nullptr, wsw2[0], g_b2[0], h);

  // ---- GIN layers 1 and 2 (din = 128) ----
  for (int l = 1; l < 3; ++l) {
    hipMemsetAsync(agg, 0, (size_t)N_NODES * HID * 4, stream);
    const int threads = N_EDGES * (HID / 4);
    scatter_add_kernel<<<(threads + 255) / 256, 256, 0, stream>>>(
        h, src, dst, agg, N_EDGES, HID);
    gemm_wmma_f32<HID, true, true><<<rowBlocks, 256, 0, stream>>>(
        h, agg, wsw1[l], g_b1[l], tbuf);
    gemm_wmma_f32<HID, false, true><<<rowBlocks, 256, 0, stream>>>(
        tbuf, nullptr, wsw2[l], g_b2[l], h);
  }

  // ---- readout: mean + max pooling per graph ----
  hipMemsetAsync(ssum, 0, (size_t)NUM_GRAPHS * HID * 4, stream);
  hipMemsetAsync(smax, 0, (size_t)NUM_GRAPHS * HID * 4, stream);  // 0 bits == 0.0f
  hipMemsetAsync(cnt,  0, (size_t)NUM_GRAPHS * 4, stream);
  {
    const int threads = N_NODES * HID;
    readout_kernel<<<(threads + 255) / 256, 256, 0, stream>>>(
        h, batch, ssum, smax, cnt, N_NODES, HID);
  }
  {
    const int threads = NUM_GRAPHS * 2 * HID;
    build_repr_kernel<<<(threads + 255) / 256, 256, 0, stream>>>(
        ssum, smax, cnt, repr, NUM_GRAPHS, HID);
  }

  // ---- head: hh = relu(repr @ head_w1 + b1); out = sigmoid(hh @ w2 + b2) ----
  gemm_wmma_f32<2 * HID, false, true><<<NUM_GRAPHS / 16, 256, 0, stream>>>(
      repr, nullptr, wswh, head_b1, hh);
  head_out_kernel<<<1, NUM_GRAPHS, 0, stream>>>(
      hh, head_w2, head_b2, out, NUM_GRAPHS, HID);
}